// CascadingSinkCacheCompile_71451075936268
// MI455X (gfx1250) — compile-verified
//
#include <hip/hip_runtime.h>

#ifndef __has_builtin
#define __has_builtin(x) 0
#endif

#if __has_builtin(__builtin_amdgcn_global_store_async_from_lds_b128)
#define HAVE_ASYNC_STORE 1
#else
#define HAVE_ASYNC_STORE 0
#endif

#if __has_builtin(__builtin_amdgcn_s_wait_asynccnt)
#define HAVE_WAIT_ASYNC 1
#else
#define HAVE_WAIT_ASYNC 0
#endif

// b128 element type matching the builtin's expected parameter
// ("int __attribute__((__vector_size__(4 * sizeof(int)))) *" per hipcc).
typedef int v4i __attribute__((vector_size(16)));
typedef __attribute__((address_space(1))) v4i* global_v4i_ptr;  // global dst
typedef __attribute__((address_space(3))) v4i* lds_v4i_ptr;     // LDS src

// Problem geometry (from reference)
constexpr int Hh      = 16;
constexpr int Tt      = 4096;
constexpr int Dd      = 128;
constexpr int MSEQ    = 8192;
constexpr int MAPPED_P = 2303;                 // positions [0,2303) hold data; rest are zero

// Flat f32 offsets in d_out (kc, vc, sc, sk, sv in return order)
constexpr long KC_ELEMS = (long)Hh * MSEQ * Dd;        // 16,777,216
constexpr long SC_OFF   = 2 * KC_ELEMS;                 // 33,554,432
constexpr long SK_OFF   = SC_OFF + MSEQ;                // 33,562,624
constexpr long SV_OFF   = SK_OFF + (long)Hh * 4 * Dd;   // 33,570,816

// float4-granularity work partition for the gather kernel
constexpr int ROW_F4     = Dd / 4;                      // 32
constexpr int PER_HEAD_A = MAPPED_P * ROW_F4;           // 73,696
constexpr int PER_CACHE_A = Hh * PER_HEAD_A;            // 1,179,136
constexpr int PART_A     = 2 * PER_CACHE_A;             // 2,358,272 (kc+vc mapped rows)
constexpr int PART_B     = MSEQ / 4;                    // 2,048 (score cache)
constexpr int PART_C     = 2 * (Hh * 4 * Dd) / 4;       // 4,096 (sk + sv)
constexpr int GATHER_TOTAL = PART_A + PART_B + PART_C;  // 2,364,416 (exact mult of 256)

// zero-fill partition
constexpr int ZROWS_PER_CHUNK = MSEQ - MAPPED_P;        // 5,889 rows per (cache,head)
constexpr int ZCHUNK  = ZROWS_PER_CHUNK * ROW_F4;       // 188,448 f4 per chunk
constexpr int ZCHUNKS = 2 * Hh;                         // 32 chunks
constexpr int ZTOTAL  = ZCHUNK * ZCHUNKS;               // 6,030,336 (exact mult of 256)

// Closed-form solution of the cascade scan: input token index stored at cache
// position p, or -1 if position is never written. tok = t_idx' + NSINK.
__device__ __forceinline__ int map_tok(int p) {
  if (p < 1020)  return 3076 + p;                 // cascade 0, 3rd wrap
  if (p < 1024)  return 2052 + p;                 // cascade 0, 2nd wrap tail
  if (p < 1534)  return 2052 + 2 * (p - 1024);    // cascade 1, evict-phase overwrites
  if (p < 2048)  return 4 + 2 * (p - 1024);       // cascade 1, surviving appends
  if (p < MAPPED_P) return 4 + 4 * (p - 2048);    // cascade 2 appends (255 tokens)
  return -1;                                      // never written -> zero
}

__global__ __launch_bounds__(256)
void cascade_gather(const float* __restrict__ keys,
                    const float* __restrict__ vals,
                    const float* __restrict__ scores,
                    float* __restrict__ out) {
  const int i = blockIdx.x * blockDim.x + threadIdx.x;
  if (i >= GATHER_TOTAL) return;

  if (i < PART_A) {
    // kc/vc mapped rows: one wave32 == one 512B row (fully coalesced b128s)
    const int cacheIdx = i / PER_CACHE_A;                 // 0 = keys, 1 = values
    int r  = i - cacheIdx * PER_CACHE_A;
    const int h  = r / PER_HEAD_A;
    r -= h * PER_HEAD_A;
    const int p  = r >> 5;        // row (cache position)
    const int q  = r & 31;        // float4 within row
    const int tok = map_tok(p);   // always >= 0 here (p < 2303)
    const float* __restrict__ src = cacheIdx ? vals : keys;
    const float4 v = *(const float4*)(src + ((long)(h * Tt + tok) * Dd + q * 4));
    float* dst = out + (long)cacheIdx * KC_ELEMS + ((long)(h * MSEQ + p) * Dd + q * 4);
    *(float4*)dst = v;
    return;
  }

  if (i < PART_A + PART_B) {
    // score cache: per-element gather (map strides are 1/2/4), b128 store
    const int j  = i - PART_A;
    const int p0 = j * 4;
    const int t0 = map_tok(p0 + 0);
    const int t1 = map_tok(p0 + 1);
    const int t2 = map_tok(p0 + 2);
    const int t3 = map_tok(p0 + 3);
    const float4 v = make_float4(t0 >= 0 ? scores[t0] : 0.0f,
                                 t1 >= 0 ? scores[t1] : 0.0f,
                                 t2 >= 0 ? scores[t2] : 0.0f,
                                 t3 >= 0 ? scores[t3] : 0.0f);
    *(float4*)(out + SC_OFF + p0) = v;
    return;
  }

  // sinks: sk[h,j] = keys[h,j]; sv[h,j] = keys[h,j] (reference stores k for both)
  const int j    = i - PART_A - PART_B;   // [0, 4096)
  const int half = j >> 11;               // 0 = sk, 1 = sv
  const int r    = j & 2047;              // f4 index inside one sink tensor
  const int h    = r >> 7;
  const int r2   = r & 127;
  const int sink = r2 >> 5;
  const int q    = r2 & 31;
  const float4 v = *(const float4*)(keys + ((long)(h * Tt + sink) * Dd + q * 4));
  float* dst = out + SK_OFF + (long)half * (Hh * 4 * Dd) + (long)r * 4;
  *(float4*)dst = v;
}

__global__ __launch_bounds__(256)
void cascade_zero(float* __restrict__ out) {
  __shared__ v4i zbuf[256];
  const int tid = threadIdx.x;
  zbuf[tid] = (v4i){0, 0, 0, 0};
  __syncthreads();

  const int i = blockIdx.x * blockDim.x + tid;
  if (i < ZTOTAL) {
    const int chunk = i / ZCHUNK;
    const int off   = i - chunk * ZCHUNK;
    const int cacheIdx = chunk >> 4;   // /16
    const int h        = chunk & 15;
    float* dst = out + (long)cacheIdx * KC_ELEMS
                     + (long)h * ((long)MSEQ * Dd)
                     + (long)MAPPED_P * Dd
                     + (long)off * 4;
#if HAVE_ASYNC_STORE
    // CDNA5 async LDS->global store: 16B per lane, 512B per wave, no VGPR data,
    // tracked by ASYNCcnt. Specific->generic implicit conversion keeps this
    // valid whether the builtin's params are AS-qualified or generic.
    __builtin_amdgcn_global_store_async_from_lds_b128(
        (global_v4i_ptr)dst,
        (lds_v4i_ptr)&zbuf[tid],
        0, 0);
#else
    *(float4*)dst = *(float4*)&zbuf[tid];
#endif
  }
#if HAVE_ASYNC_STORE && HAVE_WAIT_ASYNC
  __builtin_amdgcn_s_wait_asynccnt(0);
#endif
}

extern "C" void kernel_launch(void* const* d_in, const int* in_sizes, int n_in,
                              void* d_out, int out_size, void* d_ws, size_t ws_size,
                              hipStream_t stream) {
  const float* keys   = (const float*)d_in[0];   // (1,16,4096,128) f32
  const float* vals   = (const float*)d_in[1];   // (1,16,4096,128) f32
  const float* scores = (const float*)d_in[2];   // (4096,) f32
  float* out = (float*)d_out;                    // kc | vc | sc | sk | sv, f32

  // Disjoint writes; order between the two kernels does not matter.
  cascade_gather<<<GATHER_TOTAL / 256, 256, 0, stream>>>(keys, vals, scores, out);
  cascade_zero<<<ZTOTAL / 256, 256, 0, stream>>>(out);
}